// SCQLayer_47991964565998
// MI455X (gfx1250) — compile-verified
//
#include <hip/hip_runtime.h>
#include <math.h>

typedef float v2f __attribute__((ext_vector_type(2)));
typedef float v4f __attribute__((ext_vector_type(4)));
typedef float v8f __attribute__((ext_vector_type(8)));

#define TM 64   // block tile rows
#define TN 32   // block tile cols
#define KC 32   // k chunk
#define LDA_S 36  // padded LDS row strides (keep 16B alignment: 36*4=144B)
#define LDB_S 36

__device__ __forceinline__ void async_load_b128_to_lds(const float* gptr, void* lptr) {
    // CDNA5 async copy: 16B global -> LDS, tracked by ASYNCcnt (no VGPR round trip)
    unsigned long long ga = (unsigned long long)gptr;
    unsigned la = (unsigned)(unsigned long long)lptr;   // low 32 bits of generic ptr == LDS addr
    asm volatile("global_load_async_to_lds_b128 %0, %1, off"
                 :: "v"(la), "v"(ga) : "memory");
}
__device__ __forceinline__ void wait_async0() {
#if __has_builtin(__builtin_amdgcn_s_wait_asynccnt)
    __builtin_amdgcn_s_wait_asynccnt(0);
#else
    asm volatile("s_wait_asynccnt 0x0" ::: "memory");
#endif
}

// ---------------------------------------------------------------------------
// Generic fp32 WMMA GEMM:  D = A(MxK) * B(KxN)  [+ epilogue]
//   transB=0: B stored row-major [K][N]  -> async B128 staging
//   transB=1: B source stored as [N][K]  -> float4 load + transpose store
// mode 0: D = acc
// mode 1: D = acc + lam * (m==n)            (Gram matrix + lam*I)
// mode 2: D = Yp - 2*step*(acc - ZCp)       (FISTA pre-projection point)
// ---------------------------------------------------------------------------
__global__ __launch_bounds__(256)
void gemm_wmma_f32(const float* __restrict__ A, int lda,
                   const float* __restrict__ B, int ldb, int transB,
                   float* __restrict__ D, int ldd,
                   int M, int N, int K,
                   int mode, float lam,
                   const float* __restrict__ Yp,
                   const float* __restrict__ ZCp,
                   const float* __restrict__ stepPtr)
{
    __shared__ float As[TM][LDA_S];
    __shared__ float Bs[KC][LDB_S];

    const int t    = threadIdx.x;
    const int wave = t >> 5;      // 8 waves per block (wave32)
    const int lane = t & 31;
    const int half = lane >> 4;   // 0: K pair {0,1}, 1: K pair {2,3}
    const int l16  = lane & 15;
    const int wm   = wave & 3;    // 4 waves along M
    const int wn   = wave >> 2;   // 2 waves along N

    const int m0 = blockIdx.y * TM;
    const int n0 = blockIdx.x * TN;

    v8f acc = {0.f,0.f,0.f,0.f,0.f,0.f,0.f,0.f};

    for (int k0 = 0; k0 < K; k0 += KC) {
        // ---- stage A: 64x32 floats = 512 float4 chunks, 2 per thread (async)
        #pragma unroll
        for (int c = 0; c < 2; ++c) {
            int chunk = t + 256*c;
            int row   = chunk >> 3;          // 0..63
            int col   = (chunk & 7) << 2;    // 0,4,...,28
            async_load_b128_to_lds(&A[(size_t)(m0 + row)*lda + (k0 + col)],
                                   &As[row][col]);
        }
        // ---- stage B into Bs[k][n]
        if (!transB) {
            // 32x32 = 256 float4 chunks, 1 per thread (async, contiguous over n)
            int kr  = t >> 3;                // 0..31
            int nc  = (t & 7) << 2;          // 0..28
            async_load_b128_to_lds(&B[(size_t)(k0 + kr)*ldb + (n0 + nc)],
                                   &Bs[kr][nc]);
            wait_async0();
        } else {
            // float4 along k from source row n, transpose-store to Bs[k][n]
            int n  = t >> 3;                 // 0..31
            int k4 = (t & 7) << 2;           // 0..28
            v4f bv = *(const v4f*)&B[(size_t)(n0 + n)*ldb + (k0 + k4)];
            wait_async0();                   // A async must land before barrier too
            Bs[k4 + 0][n] = bv.x;
            Bs[k4 + 1][n] = bv.y;
            Bs[k4 + 2][n] = bv.z;
            Bs[k4 + 3][n] = bv.w;
        }
        __syncthreads();

        // ---- 8 x V_WMMA_F32_16X16X4_F32 per chunk per wave
        #pragma unroll
        for (int kk = 0; kk < KC; kk += 4) {
            const int kp = kk + 2*half;
            v2f a, b;
            // A frag 16x4: lanes 0-15 -> K={kk,kk+1}, lanes 16-31 -> K={kk+2,kk+3}
            a.x = As[16*wm + l16][kp];
            a.y = As[16*wm + l16][kp + 1];
            // B frag 4x16: same K split, N striped over lanes
            b.x = Bs[kp][16*wn + l16];
            b.y = Bs[kp + 1][16*wn + l16];
            acc = __builtin_amdgcn_wmma_f32_16x16x4_f32(
                false, a, false, b, (short)0, acc, false, false);
        }
        __syncthreads();
    }

    // C/D layout: VGPR r -> row (r + 8*half) within 16x16 tile, col = l16
    const int col = n0 + 16*wn + l16;
    const int rb  = m0 + 16*wm + 8*half;
    if (mode == 0) {
        #pragma unroll
        for (int r = 0; r < 8; ++r)
            D[(size_t)(rb + r)*ldd + col] = acc[r];
    } else if (mode == 1) {
        #pragma unroll
        for (int r = 0; r < 8; ++r) {
            int m = rb + r;
            D[(size_t)m*ldd + col] = acc[r] + ((m == col) ? lam : 0.f);
        }
    } else {
        const float step = *stepPtr;
        #pragma unroll
        for (int r = 0; r < 8; ++r) {
            size_t idx = (size_t)(rb + r)*ldd + col;
            D[idx] = Yp[idx] - 2.f*step*(acc[r] - ZCp[idx]);
        }
    }
}

// ---------------------------------------------------------------------------
// Reductions (wave32-aware)
// ---------------------------------------------------------------------------
__device__ inline float waveReduceSum(float x) {
    #pragma unroll
    for (int off = 16; off > 0; off >>= 1)
        x += __shfl_down(x, off, 32);
    return x;
}
__device__ inline float waveReduceMax(float x) {
    #pragma unroll
    for (int off = 16; off > 0; off >>= 1)
        x = fmaxf(x, __shfl_down(x, off, 32));
    return x;
}
// 256-thread block reductions; sred is shared float[9]
__device__ inline float blockReduceSum256(float x, float* sred) {
    int t = threadIdx.x, w = t >> 5, lane = t & 31;
    float s = waveReduceSum(x);
    __syncthreads();
    if (lane == 0) sred[w] = s;
    __syncthreads();
    if (t == 0) {
        float tot = 0.f;
        #pragma unroll
        for (int i = 0; i < 8; ++i) tot += sred[i];
        sred[8] = tot;
    }
    __syncthreads();
    return sred[8];
}
__device__ inline float blockReduceMax256(float x, float* sred) {
    int t = threadIdx.x, w = t >> 5, lane = t & 31;
    float s = waveReduceMax(x);
    __syncthreads();
    if (lane == 0) sred[w] = s;
    __syncthreads();
    if (t == 0) {
        float tot = sred[0];
        #pragma unroll
        for (int i = 1; i < 8; ++i) tot = fmaxf(tot, sred[i]);
        sred[8] = tot;
    }
    __syncthreads();
    return sred[8];
}

// ---------------------------------------------------------------------------
// Power iteration on G (1024x1024), single workgroup; step = 1/(2 v^T G v)
// Coalesced: acc_t = sum_j G[j][t] * v[j]  (G symmetric)
// ---------------------------------------------------------------------------
__global__ __launch_bounds__(1024)
void power_iter_kernel(const float* __restrict__ G, float* __restrict__ stepOut)
{
    __shared__ float v[1024];
    __shared__ float red[33];
    const int t = threadIdx.x, w = t >> 5, lane = t & 31;
    v[t] = 1.0f / 32.0f;          // ones(1024)/sqrt(1024)
    __syncthreads();

    float acc = 0.f;
    for (int it = 0; it < 20; ++it) {
        acc = 0.f;
        for (int j = 0; j < 1024; ++j)
            acc += G[(size_t)j*1024 + t] * v[j];
        float s = waveReduceSum(acc * acc);
        if (lane == 0) red[w] = s;
        __syncthreads();
        if (w == 0) {
            float x = red[lane];
            x = waveReduceSum(x);
            if (lane == 0) red[32] = sqrtf(x) + 1e-12f;
        }
        __syncthreads();
        const float nrm = red[32];
        __syncthreads();
        v[t] = acc / nrm;
        __syncthreads();
    }
    // L = 2 * v^T G v with normalized v
    acc = 0.f;
    for (int j = 0; j < 1024; ++j)
        acc += G[(size_t)j*1024 + t] * v[j];
    float s = waveReduceSum(v[t] * acc);
    if (lane == 0) red[w] = s;
    __syncthreads();
    if (t == 0) {
        float tot = 0.f;
        for (int i = 0; i < 32; ++i) tot += red[i];
        stepOut[0] = 1.0f / (2.0f * tot);
    }
}

// ---------------------------------------------------------------------------
// Per-row exact simplex projection (bisection + exact active-set refinement)
// + FISTA momentum update. One block (256 threads) per row, K=1024.
// Thread t owns the 4 consecutive elements [4t, 4t+4): projection math is
// order-invariant across the row, so float4 (B128) I/O is legal.
// ---------------------------------------------------------------------------
__global__ __launch_bounds__(256)
void fista_update_kernel(const float* __restrict__ V, float* __restrict__ alpha,
                         float* __restrict__ Y, float mom)
{
    __shared__ float sred[9];
    const int row = blockIdx.x, t = threadIdx.x;
    const size_t base = (size_t)row * 1024 + 4*t;
    v4f v = *(const v4f*)(V + base);

    float mx = fmaxf(fmaxf(v.x, v.y), fmaxf(v.z, v.w));
    mx = blockReduceMax256(mx, sred);

    // theta* in [mx-1, mx]; f(theta)=sum(max(v-theta,0))-1 decreasing
    float lo = mx - 1.0f, hi = mx;
    for (int it = 0; it < 48; ++it) {
        float mid = 0.5f * (lo + hi);
        float p = fmaxf(v.x - mid, 0.f) + fmaxf(v.y - mid, 0.f)
                + fmaxf(v.z - mid, 0.f) + fmaxf(v.w - mid, 0.f);
        float s = blockReduceSum256(p, sred);
        if (s > 1.0f) lo = mid; else hi = mid;   // uniform branch
    }
    const float th0 = 0.5f * (lo + hi);
    // exact theta on identified active set
    float ps = 0.f, pn = 0.f;
    #pragma unroll
    for (int i = 0; i < 4; ++i) {
        float vi = v[i];
        if (vi > th0) { ps += vi; pn += 1.f; }
    }
    float S = blockReduceSum256(ps, sred);
    float n = blockReduceSum256(pn, sred);
    n = fmaxf(n, 1.f);
    const float theta = (S - 1.0f) / n;

    v4f old = *(const v4f*)(alpha + base);
    v4f a, ynew;
    #pragma unroll
    for (int i = 0; i < 4; ++i) {
        a[i]    = fmaxf(v[i] - theta, 0.f);
        ynew[i] = a[i] + mom * (a[i] - old[i]);
    }
    *(v4f*)(alpha + base) = a;
    *(v4f*)(Y + base)     = ynew;
}

__global__ __launch_bounds__(256)
void init_kernel(float* __restrict__ alpha, float* __restrict__ Y)
{
    size_t i = ((size_t)blockIdx.x * 256 + threadIdx.x) * 4;
    const v4f u = {1.0f/1024.0f, 1.0f/1024.0f, 1.0f/1024.0f, 1.0f/1024.0f};
    *(v4f*)(alpha + i) = u;
    *(v4f*)(Y + i)     = u;
}

// entropy (mean of -sum a_safe log a_safe) and sparsity (mean L1)
__global__ __launch_bounds__(256)
void stats_kernel(const float* __restrict__ alpha,
                  float* __restrict__ entropy, float* __restrict__ sparsity)
{
    __shared__ float sred[9];
    const int row = blockIdx.x, t = threadIdx.x;
    v4f a = *(const v4f*)(alpha + (size_t)row * 1024 + 4*t);
    float esum = 0.f, ssum = 0.f;
    #pragma unroll
    for (int i = 0; i < 4; ++i) {
        float as = a[i] + 1e-10f;
        esum += -as * logf(as);
        ssum += fabsf(a[i]);
    }
    float e = blockReduceSum256(esum, sred);
    float s = blockReduceSum256(ssum, sred);
    if (t == 0) {
        atomicAdd(entropy,  e * (1.0f / 1024.0f));
        atomicAdd(sparsity, s * (1.0f / 1024.0f));
    }
}

// num_active = sum over (h,w,k) of any_b( alpha4[b,h,w,k] > 1e-3 )
// 65536 float4 groups: idx -> hw = idx>>8, k = (idx&255)*4
__global__ __launch_bounds__(256)
void active_kernel(const float* __restrict__ alpha, float* __restrict__ num_active)
{
    __shared__ float sred[9];
    const int idx = blockIdx.x * 256 + threadIdx.x;   // 0..65535
    const int hw  = idx >> 8;                         // 0..255  (h*16+w)
    const int k   = (idx & 255) << 2;
    v4f m = {0.f, 0.f, 0.f, 0.f};
    #pragma unroll
    for (int b = 0; b < 4; ++b) {
        v4f a = *(const v4f*)(alpha + (size_t)(hw + 256*b) * 1024 + k);
        #pragma unroll
        for (int i = 0; i < 4; ++i)
            m[i] = fmaxf(m[i], a[i]);
    }
    float cnt = 0.f;
    #pragma unroll
    for (int i = 0; i < 4; ++i)
        cnt += (m[i] > 1e-3f) ? 1.f : 0.f;
    float c = blockReduceSum256(cnt, sred);
    if (threadIdx.x == 0) atomicAdd(num_active, c);
}

// ---------------------------------------------------------------------------
extern "C" void kernel_launch(void* const* d_in, const int* in_sizes, int n_in,
                              void* d_out, int out_size, void* d_ws, size_t ws_size,
                              hipStream_t stream)
{
    const float* z  = (const float*)d_in[0];   // (4,16,16,256) == z_flat (1024,256)
    const float* cb = (const float*)d_in[1];   // (1024,256)

    float* out   = (float*)d_out;
    float* zq    = out;                                    // 1024*256
    float* alpha = out + 1024*256;                         // 1024*1024 (alpha4 output)
    float* scal  = out + 1024*256 + 1024*1024;             // entropy, sparsity, num_active

    float*       ws = (float*)d_ws;
    const size_t MM = 1024ull * 1024ull;
    float* G       = ws;
    float* ZC      = ws + 1*MM;
    float* Y       = ws + 2*MM;
    float* V       = ws + 3*MM;
    float* stepPtr = ws + 4*MM;

    hipMemsetAsync(scal, 0, 3 * sizeof(float), stream);

    const dim3 blk(256);
    const dim3 grid_sq(1024 / TN, 1024 / TM);

    // G = C C^T + lam I   (B[n][k] = cb[n][k] -> transB=1)
    gemm_wmma_f32<<<grid_sq, blk, 0, stream>>>(
        cb, 256, cb, 256, 1, G, 1024, 1024, 1024, 256,
        1, 1e-3f, nullptr, nullptr, nullptr);
    // ZC = Z C^T
    gemm_wmma_f32<<<grid_sq, blk, 0, stream>>>(
        z, 256, cb, 256, 1, ZC, 1024, 1024, 1024, 256,
        0, 0.f, nullptr, nullptr, nullptr);

    init_kernel<<<1024, blk, 0, stream>>>(alpha, Y);
    power_iter_kernel<<<1, 1024, 0, stream>>>(G, stepPtr);

    // FISTA: t-sequence is a pure scalar recurrence -> host precomputed momentum
    double tcur = 1.0;
    for (int it = 0; it < 80; ++it) {
        double tnext = 0.5 * (1.0 + sqrt(1.0 + 4.0 * tcur * tcur));
        float  mom   = (float)((tcur - 1.0) / tnext);
        tcur = tnext;
        // V = Y - 2*step*(Y@G - ZC)   (G symmetric, row-major, transB=0)
        gemm_wmma_f32<<<grid_sq, blk, 0, stream>>>(
            Y, 1024, G, 1024, 0, V, 1024, 1024, 1024, 1024,
            2, 0.f, Y, ZC, stepPtr);
        fista_update_kernel<<<1024, blk, 0, stream>>>(V, alpha, Y, mom);
    }

    // z_q = alpha @ cb   (M=1024, N=256, K=1024)
    gemm_wmma_f32<<<dim3(256 / TN, 1024 / TM), blk, 0, stream>>>(
        alpha, 1024, cb, 256, 0, zq, 256, 1024, 256, 1024,
        0, 0.f, nullptr, nullptr, nullptr);

    stats_kernel<<<1024, blk, 0, stream>>>(alpha, scal + 0, scal + 1);
    active_kernel<<<256, blk, 0, stream>>>(alpha, scal + 2);
}